// N3Tree_16587163697588
// MI455X (gfx1250) — compile-verified
//
#include <hip/hip_runtime.h>
#include <stdint.h>

// N3-tree query: traverse octree per lane, then wave-cooperative transposed
// payload copy through LDS using CDNA5 async global<->LDS DMA ops.
// Output stores use TH_STORE_NT: the 128MB result is write-once streaming,
// while the gathered data/child tables stay L2-resident.

__global__ __launch_bounds__(256) void n3tree_query_kernel(
    const float* __restrict__ indices,   // [Q][3]
    const float* __restrict__ data,      // [n_nodes][2][2][2][32]
    const int*   __restrict__ child,     // [n_nodes][2][2][2]
    const float* __restrict__ offset,    // [3]
    const float* __restrict__ invradius, // [1]
    float* __restrict__ out,             // [Q][32]
    int Q)
{
    // 8 waves/block * 4KB staging tile per wave
    __shared__ __align__(16) float smem[8192];

    const int tid      = threadIdx.x;
    const int lane     = tid & 31;
    const int wave     = tid >> 5;
    const int q        = blockIdx.x * blockDim.x + tid;
    const int tileBase = q - lane;            // first query owned by this wave
    if (tileBase >= Q) return;
    const int qc = (q < Q) ? q : (Q - 1);

    // ---- uniform scalars ----
    const float ir = invradius[0];
    const float ox = offset[0], oy = offset[1], oz = offset[2];

    // jnp: clip(t, 0, 1-1e-10); in f32 the upper bound rounds to 1.0f,
    // and min(floor(2t),1) handles t==1 exactly like the reference.
    const float hi = 1.0f - 1e-10f;
    float tx = fminf(fmaxf(indices[3 * qc + 0] * ir + ox, 0.0f), hi);
    float ty = fminf(fmaxf(indices[3 * qc + 1] * ir + oy, 0.0f), hi);
    float tz = fminf(fmaxf(indices[3 * qc + 2] * ir + oz, 0.0f), hi);

    // ---- branchless traversal, MAX_LEVELS = 7 ----
    int node = 0;
    int leaf = 0;          // default node 0, cell (0,0,0) as in reference
    bool done = false;
#pragma unroll
    for (int l = 0; l < 7; ++l) {
        const float sx = tx * 2.0f, sy = ty * 2.0f, sz = tz * 2.0f;
        const float fx = fminf(floorf(sx), 1.0f);
        const float fy = fminf(floorf(sy), 1.0f);
        const float fz = fminf(floorf(sz), 1.0f);
        const int cell = (((int)fx) << 2) | (((int)fy) << 1) | ((int)fz);
        const int idx  = node * 8 + cell;
        const int delta = child[idx];
        const bool term = (delta == 0) && !done;
        const bool adv  = (delta != 0) && !done;
        leaf = term ? idx : leaf;
        node = adv ? node + delta : node;
        tx   = adv ? sx - fx : tx;
        ty   = adv ? sy - fy : ty;
        tz   = adv ? sz - fz : tz;
        done = done || term;
    }

    const int leafByteOff = leaf << 7;  // (node*8+cell) * 32 floats * 4B
    // Warm L2 for the gather phase (CDNA5 global_prefetch_b8)
    __builtin_prefetch((const char*)data + leafByteOff, 0, 0);

    // LDS byte address of this wave's 4KB tile (generic ptr low 32 bits == LDS offset)
    const unsigned smemBase = (unsigned)(uintptr_t)(&smem[0]) + ((unsigned)wave << 12);
    const unsigned sub = (unsigned)lane & 7u;   // 16B chunk within a 128B row

    // ---- Phase 1: async gather of 32 leaf rows into LDS ----
    // Instruction i covers queries 4i..4i+3: 4 rows x 8 lanes x 16B = 512B.
#pragma unroll
    for (int i = 0; i < 8; ++i) {
        const int qsel = (i << 2) | (lane >> 3);
        const int off  = __shfl(leafByteOff, qsel, 32);   // wave32 broadcast
        const char* gsrc = (const char*)data + off + sub * 16u;
        const unsigned ldsDst = smemBase + ((unsigned)qsel << 7) + sub * 16u;
        asm volatile("global_load_async_to_lds_b128 %0, %1, off"
                     :: "v"(ldsDst), "v"(gsrc) : "memory");
    }
    asm volatile("s_wait_asynccnt 0x0" ::: "memory");

    // ---- Phase 2: async stream the 4KB tile to contiguous output (NT) ----
#pragma unroll
    for (int i = 0; i < 8; ++i) {
        const unsigned p = ((unsigned)i << 9) + (unsigned)lane * 16u;  // byte in tile
        if (tileBase + (int)(p >> 7) < Q) {
            char* gdst = (char*)out + (size_t)tileBase * 128u + p;
            const unsigned ldsSrc = smemBase + p;
            asm volatile("global_store_async_from_lds_b128 %0, %1, off th:TH_STORE_NT"
                         :: "v"(gdst), "v"(ldsSrc) : "memory");
        }
    }
    asm volatile("s_wait_asynccnt 0x0" ::: "memory");
}

extern "C" void kernel_launch(void* const* d_in, const int* in_sizes, int n_in,
                              void* d_out, int out_size, void* d_ws, size_t ws_size,
                              hipStream_t stream) {
    const float* indices  = (const float*)d_in[0];
    const float* data     = (const float*)d_in[1];
    const int*   child    = (const int*)d_in[2];
    const float* offset   = (const float*)d_in[3];
    const float* invrad   = (const float*)d_in[4];
    float* out = (float*)d_out;

    const int Q = in_sizes[0] / 3;
    const int grid = (Q + 255) / 256;
    n3tree_query_kernel<<<grid, 256, 0, stream>>>(indices, data, child, offset,
                                                  invrad, out, Q);
}